// NeuralSDETrajectoryPredictor_22265110462778
// MI455X (gfx1250) — compile-verified
//
#include <hip/hip_runtime.h>
#include <stdint.h>

typedef float v2f __attribute__((ext_vector_type(2)));
typedef float v8f __attribute__((ext_vector_type(8)));

#define HID     128
#define CTX     256
#define NS      8
#define NB      16
#define NO      64
#define NPER    1024          // b*o
#define NROWS   8192          // ns*b*o
#define HORIZON 40
#define TM      128           // rows per block in step kernel
#define HP      132           // LDS stride (floats) for sH/sG: A-frag conflict-free
#define WROW    288           // LDS stride (floats) per K-pair row of packed W
                              // (288 mod 64 == 32 -> half-waves in disjoint banks)

#define MEAN_ELEMS (NB*HORIZON*NO*2)   // 81920
#define PATH_ELEMS (NS*MEAN_ELEMS)     // 655360

__device__ __forceinline__ float gelu_exact(float x) {
    return 0.5f * x * (1.0f + erff(x * 0.70710678118654752f));
}

// One 16-byte async copy global->LDS per lane (GVS mode, ASYNCcnt-tracked).
__device__ __forceinline__ void async_copy_b128(uint32_t lds_addr,
                                                const float* gbase,
                                                uint32_t byte_off) {
    asm volatile("global_load_async_to_lds_b128 %0, %1, %2 offset:0"
                 :: "v"(lds_addr), "v"(byte_off), "s"(gbase) : "memory");
}
__device__ __forceinline__ void wait_async0() {
    asm volatile("s_wait_asynccnt 0" ::: "memory");
}

// ctxp16[bi][j] = bp[j] + sum_k z[bi][k] * Wp[2+k][j]   (16x256 @ 256x128, tiny)
__global__ void ctx_proj_kernel(const float* __restrict__ z,
                                const float* __restrict__ Wp,
                                const float* __restrict__ bp,
                                float* __restrict__ ctxp) {
    const int bi = blockIdx.x;
    const int j  = threadIdx.x;
    float acc = bp[j];
    const float* zrow = z + bi * CTX;
    for (int k = 0; k < CTX; ++k)
        acc = fmaf(zrow[k], Wp[(2 + k) * HID + j], acc);
    ctxp[bi * HID + j] = acc;
}

// state0[r] = traj_hist[bi, T-1, oi, :], replicated over samples s
__global__ void state_init_kernel(const float* __restrict__ traj_hist,
                                  float* __restrict__ state) {
    const int r = blockIdx.x * blockDim.x + threadIdx.x;
    if (r >= NROWS) return;
    const int n  = r & (NPER - 1);
    const int bi = n >> 6, oi = n & 63;
    const int src = ((bi * 12 + 11) * 64 + oi) * 2;
    state[r * 2 + 0] = traj_hist[src + 0];
    state[r * 2 + 1] = traj_hist[src + 1];
}

// K-pair interleave Wd1/Ws1 once: P[kp*256 + n*2 + h] = W[(2kp+h)*128 + n].
// Makes each WMMA B-fragment one contiguous aligned 8-byte LDS load.
__global__ void pack_weights_kernel(const float* __restrict__ Wd1,
                                    const float* __restrict__ Ws1,
                                    float* __restrict__ Wd1p,
                                    float* __restrict__ Ws1p) {
    const int idx = blockIdx.x * blockDim.x + threadIdx.x;   // 0..32767
    const int e   = idx & 16383;
    const int kp  = e >> 8;
    const int q   = e & 255;
    const int n   = q >> 1;
    const int h   = q & 1;
    const float* src = (idx >> 14) ? Ws1 : Wd1;
    float*       dst = (idx >> 14) ? Ws1p : Wd1p;
    dst[e] = src[(2 * kp + h) * HID + n];
}

__global__ __launch_bounds__(256) void sde_step_kernel(
    const float* __restrict__ noise,   // [H, ns, N, 2]
    const float* __restrict__ Wp,      // [258,128] (rows 0,1 used here)
    const float* __restrict__ Wd1p, const float* __restrict__ bd1,
    const float* __restrict__ Wd2,  const float* __restrict__ bd2,
    const float* __restrict__ Ws1p, const float* __restrict__ bs1,
    const float* __restrict__ Ws2,  const float* __restrict__ bs2,
    const float* __restrict__ ctxp,    // [16,128]
    float* __restrict__ state,         // [8192,2] in workspace
    float* __restrict__ paths_out,     // d_out + 81920
    float* __restrict__ sig_out,       // d_out + 737280
    int t)
{
    __shared__ __align__(16) float sWa[64 * WROW];  // packed Wd1 (async-staged)
    __shared__ __align__(16) float sWb[64 * WROW];  // packed Ws1 (async-staged)
    __shared__ __align__(16) float sH[TM * HP];     // h activations
    __shared__ __align__(16) float sG[TM * HP];     // gelu(GEMM) output
    __shared__ float sWd2[HID * 2];
    __shared__ float sWs2[HID * 2];
    __shared__ float sDrift[TM * 2];
    __shared__ float sSig[TM * 2];

    const int tid   = threadIdx.x;
    const int lane  = tid & 31;
    const int wid   = tid >> 5;        // 8 waves
    const int lmod  = lane & 15;
    const int lhalf = lane >> 4;
    const int row0  = blockIdx.x * TM;

    const uint32_t sWa_base = (uint32_t)(uintptr_t)(&sWa[0]);
    const uint32_t sWb_base = (uint32_t)(uintptr_t)(&sWb[0]);

    // ---- issue async stages of both packed weight matrices (double buffer) ----
    #pragma unroll
    for (int i = 0; i < 16; ++i) {
        const int idx4 = i * 256 + tid;            // 0..4095 b128 chunks
        const int kp = idx4 >> 6;                  // 64 chunks per packed row
        const int c  = idx4 & 63;
        const uint32_t lds_off = (uint32_t)(kp * WROW * 4 + c * 16);
        async_copy_b128(sWa_base + lds_off, Wd1p, (uint32_t)idx4 * 16u);
        async_copy_b128(sWb_base + lds_off, Ws1p, (uint32_t)idx4 * 16u);
    }
    // small projection weights -> LDS (256 floats each)
    sWd2[tid] = Wd2[tid];
    sWs2[tid] = Ws2[tid];

    // ---- h = state @ Wp[0:2] + ctxp16[b]  (rank-2 update, bias pre-folded) ----
    {
        const int m  = tid >> 1;
        const int c0 = (tid & 1) * 64;
        const int r  = row0 + m;
        const int n  = r & (NPER - 1);
        const int bi = n >> 6;
        const float s0 = state[r * 2 + 0], s1 = state[r * 2 + 1];
        const float* wp0 = Wp;
        const float* wp1 = Wp + HID;
        const float* cp  = ctxp + bi * HID;
        for (int j = c0; j < c0 + 64; ++j)
            sH[m * HP + j] = fmaf(s0, wp0[j], fmaf(s1, wp1[j], cp[j]));
    }
    wait_async0();
    __syncthreads();

    // ---- GEMM1: gelu(h @ Wd1 + bd1) -> sG, via V_WMMA_F32_16X16X4_F32 ----
    {
        v8f acc[8];
        #pragma unroll
        for (int nt = 0; nt < 8; ++nt) {
            const float bv = bd1[nt * 16 + lmod];
            #pragma unroll
            for (int i = 0; i < 8; ++i) acc[nt][i] = bv;
        }
        const int mbase = wid * 16;
        for (int kc = 0; kc < 32; ++kc) {
            const int kb    = kc * 4 + 2 * lhalf;    // half-wave K split (ISA layout)
            const int kpair = kc * 2 + lhalf;
            const v2f a = *(const v2f*)&sH[(mbase + lmod) * HP + kb];        // ds_load_b64
            #pragma unroll
            for (int nt = 0; nt < 8; ++nt) {
                const v2f b = *(const v2f*)&sWa[kpair * WROW + (nt * 16 + lmod) * 2];
                acc[nt] = __builtin_amdgcn_wmma_f32_16x16x4_f32(
                    false, a, false, b, (short)0, acc[nt], false, false);
            }
        }
        #pragma unroll
        for (int nt = 0; nt < 8; ++nt)
            #pragma unroll
            for (int i = 0; i < 8; ++i)
                sG[(mbase + i + 8 * lhalf) * HP + nt * 16 + lmod] = gelu_exact(acc[nt][i]);
    }
    __syncthreads();

    // ---- drift = g1 @ Wd2 + bd2 ----
    {
        const int m = wid * 16 + (lane >> 1);
        const int c = lane & 1;
        float acc = bd2[c];
        const float* g = sG + m * HP;
        for (int j = 0; j < HID; ++j)
            acc = fmaf(g[j], sWd2[j * 2 + c], acc);
        sDrift[m * 2 + c] = acc;
    }
    __syncthreads();

    // ---- GEMM2: gelu(h @ Ws1 + bs1) -> sG ----
    {
        v8f acc[8];
        #pragma unroll
        for (int nt = 0; nt < 8; ++nt) {
            const float bv = bs1[nt * 16 + lmod];
            #pragma unroll
            for (int i = 0; i < 8; ++i) acc[nt][i] = bv;
        }
        const int mbase = wid * 16;
        for (int kc = 0; kc < 32; ++kc) {
            const int kb    = kc * 4 + 2 * lhalf;
            const int kpair = kc * 2 + lhalf;
            const v2f a = *(const v2f*)&sH[(mbase + lmod) * HP + kb];
            #pragma unroll
            for (int nt = 0; nt < 8; ++nt) {
                const v2f b = *(const v2f*)&sWb[kpair * WROW + (nt * 16 + lmod) * 2];
                acc[nt] = __builtin_amdgcn_wmma_f32_16x16x4_f32(
                    false, a, false, b, (short)0, acc[nt], false, false);
            }
        }
        #pragma unroll
        for (int nt = 0; nt < 8; ++nt)
            #pragma unroll
            for (int i = 0; i < 8; ++i)
                sG[(mbase + i + 8 * lhalf) * HP + nt * 16 + lmod] = gelu_exact(acc[nt][i]);
    }
    __syncthreads();

    // ---- sigma = D_MIN + (D_MAX-D_MIN)*sigmoid(g2 @ Ws2 + bs2) ----
    {
        const int m = wid * 16 + (lane >> 1);
        const int c = lane & 1;
        float acc = bs2[c];
        const float* g = sG + m * HP;
        for (int j = 0; j < HID; ++j)
            acc = fmaf(g[j], sWs2[j * 2 + c], acc);
        sSig[m * 2 + c] = 0.05f + 1.95f / (1.0f + expf(-acc));
    }
    __syncthreads();

    // ---- SDE update: delta clamp (tanh), geo clamp, write state/paths/sig ----
    {
        const int m = tid >> 1;
        const int c = tid & 1;
        const int r = row0 + m;
        const float sg  = sSig[m * 2 + c];
        const float noi = noise[(t * NROWS + r) * 2 + c];
        float delta = sDrift[m * 2 + c] + sg * noi;      // DT = 1
        delta = 5.0f * tanhf(delta * 0.2f);              // MAX_STEP_DEG = 5
        float x = state[r * 2 + c] + delta;
        if (c == 0) {
            x = fminf(fmaxf(x, -90.0f), 90.0f);
        } else {
            x = fmodf(x, 360.0f);
            if (x < 0.0f) x += 360.0f;
        }
        state[r * 2 + c] = x;
        const int s  = r >> 10;
        const int n  = r & (NPER - 1);
        const int bi = n >> 6, oi = n & 63;
        const int pidx = ((s * NB + bi) * HORIZON + t) * (NO * 2) + oi * 2 + c;
        paths_out[pidx] = x;
        sig_out[pidx]   = sg;
    }
}

// traj_mean = mean over ns of paths
__global__ void mean_kernel(const float* __restrict__ paths,
                            float* __restrict__ mean) {
    const int idx = blockIdx.x * blockDim.x + threadIdx.x;
    if (idx >= MEAN_ELEMS) return;
    float s = 0.0f;
    #pragma unroll
    for (int k = 0; k < NS; ++k) s += paths[k * MEAN_ELEMS + idx];
    mean[idx] = s * 0.125f;
}

extern "C" void kernel_launch(void* const* d_in, const int* in_sizes, int n_in,
                              void* d_out, int out_size, void* d_ws, size_t ws_size,
                              hipStream_t stream) {
    const float* z   = (const float*)d_in[0];
    const float* th  = (const float*)d_in[1];
    const float* noi = (const float*)d_in[2];
    const float* Wp  = (const float*)d_in[3];
    const float* bp  = (const float*)d_in[4];
    const float* Wd1 = (const float*)d_in[5];
    const float* bd1 = (const float*)d_in[6];
    const float* Wd2 = (const float*)d_in[7];
    const float* bd2 = (const float*)d_in[8];
    const float* Ws1 = (const float*)d_in[9];
    const float* bs1 = (const float*)d_in[10];
    const float* Ws2 = (const float*)d_in[11];
    const float* bs2 = (const float*)d_in[12];
    (void)in_sizes; (void)n_in; (void)out_size; (void)ws_size;

    float* out   = (float*)d_out;
    float* mean  = out;                       // [16,40,64,2]
    float* paths = out + MEAN_ELEMS;          // [8,16,40,64,2]
    float* sig   = out + MEAN_ELEMS + PATH_ELEMS;

    float* state = (float*)d_ws;              // [8192,2]
    float* ctxp  = state + NROWS * 2;         // [16,128]
    float* Wd1p  = ctxp + NB * HID;           // [64,128] float2 packed
    float* Ws1p  = Wd1p + HID * HID;          // [64,128] float2 packed

    ctx_proj_kernel<<<NB, HID, 0, stream>>>(z, Wp, bp, ctxp);
    state_init_kernel<<<NROWS / 256, 256, 0, stream>>>(th, state);
    pack_weights_kernel<<<(2 * HID * HID) / 256, 256, 0, stream>>>(Wd1, Ws1, Wd1p, Ws1p);

    for (int t = 0; t < HORIZON; ++t) {
        sde_step_kernel<<<NROWS / TM, 256, 0, stream>>>(
            noi, Wp, Wd1p, bd1, Wd2, bd2, Ws1p, bs1, Ws2, bs2,
            ctxp, state, paths, sig, t);
    }

    mean_kernel<<<(MEAN_ELEMS + 255) / 256, 256, 0, stream>>>(paths, mean);
}